// Model_17111149707390
// MI455X (gfx1250) — compile-verified
//
#include <hip/hip_runtime.h>
#include <stdint.h>
#include <stddef.h>

#define TPW 8            // tokens per wave (grid-stride chunk)
#define WAVES_PER_BLOCK 8

typedef __attribute__((ext_vector_type(4))) float v4f;   // clang-native vec4 (NT-store capable)

// sigmoid(v) = 1/(1+exp(-v)) via v_exp_f32 + v_rcp_f32 (+1 Newton step)
__device__ __forceinline__ float fast_sigmoid(float v) {
    float e = __builtin_amdgcn_exp2f(v * -1.44269504088896340736f); // exp(-v)
    float d = 1.0f + e;
    float r = __builtin_amdgcn_rcpf(d);
    r = __builtin_fmaf(__builtin_fmaf(-d, r, 1.0f), r, r);          // NR refine
    return r;
}

__device__ __forceinline__ unsigned umax32(unsigned a, unsigned b) { return a > b ? a : b; }
__device__ __forceinline__ unsigned umin32(unsigned a, unsigned b) { return a < b ? a : b; }

// monotone float -> u32 map (order-preserving for all finite values)
__device__ __forceinline__ unsigned fsortable(float v) {
    unsigned u = __float_as_uint(v);
    return u ^ ((unsigned)((int)u >> 31) | 0x80000000u);
}

// descending compare-exchange: a := max, b := min
#define CE(a, b) { unsigned _hi = umax32(a, b), _lo = umin32(a, b); a = _hi; b = _lo; }

// running top-2 update with value v
#define T2(v) { float _nm = fmaxf(m1, v); m2 = fmaxf(m2, fminf(m1, v)); m1 = _nm; }

__global__ __launch_bounds__(256) void moe_gate_topk(
    const float* __restrict__ x, const float* __restrict__ bias,
    float* __restrict__ out_y, int* __restrict__ out_idx,
    float* __restrict__ out_scores, int n)
{
    const int lane = (int)(threadIdx.x & 31u);
    const int wv   = (int)(threadIdx.x >> 5);
    const int myg  = lane >> 2;                      // my expert group (0..7)
    const int t0   = (blockIdx.x * WAVES_PER_BLOCK + wv) * TPW;
    if (t0 >= n) return;

    const unsigned laneoff = (unsigned)(lane * 8);   // element offset of my 8 experts

    // bias for this lane's 8 contiguous experts (tiny, L0-resident)
    const v4f* bp = (const v4f*)(bias + laneoff);
    const v4f b0 = bp[0];
    const v4f b1 = bp[1];

    // software pipeline: preload token t0's row slice (2 x b128 per lane)
    {
        // fallthrough into loop body with cx preloaded
    }
    const v4f* xr = (const v4f*)(x + (((unsigned)t0 << 8) | laneoff));
    v4f cx0 = xr[0];
    v4f cx1 = xr[1];

    for (int i = 0; i < TPW; ++i) {
        const int t = t0 + i;
        if (t >= n) break;
        const unsigned rowoff = ((unsigned)t << 8) | laneoff;   // 32-bit element offset
        const v4f x0 = cx0, x1 = cx1;
        if (i + 1 < TPW && t + 1 < n) {
            const v4f* nx = (const v4f*)(x + (rowoff + 256u));
            cx0 = nx[0];
            cx1 = nx[1];
            if (i + 2 < TPW && t + 2 < n) {
                // gfx1250 global_prefetch_b8: pull token t+2 toward L2/L0
                __builtin_prefetch(x + (rowoff + 512u), 0, 3);
            }
        }

        // --- sigmoid scores -------------------------------------------------
        const float s0 = fast_sigmoid(x0.x), s1 = fast_sigmoid(x0.y);
        const float s2 = fast_sigmoid(x0.z), s3 = fast_sigmoid(x0.w);
        const float s4 = fast_sigmoid(x1.x), s5 = fast_sigmoid(x1.y);
        const float s6 = fast_sigmoid(x1.z), s7 = fast_sigmoid(x1.w);

        // stream scores out nontemporally (134 MB, write-once)
        {
            v4f sv0 = { s0, s1, s2, s3 };
            v4f sv1 = { s4, s5, s6, s7 };
            v4f* sp = (v4f*)(out_scores + rowoff);
            __builtin_nontemporal_store(sv0, sp);
            __builtin_nontemporal_store(sv1, sp + 1);
        }

        // biased scores (selection metric)
        const float e0 = s0 + b0.x, e1 = s1 + b0.y, e2 = s2 + b0.z, e3 = s3 + b0.w;
        const float e4 = s4 + b1.x, e5 = s5 + b1.y, e6 = s6 + b1.z, e7 = s7 + b1.w;

        // --- group score = sum of top-2 within group (32 experts = 4 lanes) -
        float m1 = fmaxf(e0, e1), m2 = fminf(e0, e1);
        T2(e2) T2(e3) T2(e4) T2(e5) T2(e6) T2(e7)
        {
            float o1 = __shfl_xor(m1, 1, 32), o2 = __shfl_xor(m2, 1, 32);
            float nm = fmaxf(m1, o1);
            m2 = fmaxf(fmaxf(m2, o2), fminf(m1, o1)); m1 = nm;
        }
        {
            float o1 = __shfl_xor(m1, 2, 32), o2 = __shfl_xor(m2, 2, 32);
            float nm = fmaxf(m1, o1);
            m2 = fmaxf(fmaxf(m2, o2), fminf(m1, o1)); m1 = nm;
        }
        const float gscore = m1 + m2;   // identical across the 4 lanes of a group

        // --- rank my group among the 8; keep top-4 (tie -> lower index) -----
        // packed group key: sortable score (top 29b) | (7 - group) -> strict u32 order
        const unsigned gk = (fsortable(gscore) & 0xFFFFFFF8u) | (unsigned)(7 - myg);
        int rank = 0;
        #pragma unroll
        for (int h = 0; h < 8; ++h) {
            unsigned gh = (unsigned)__shfl((int)gk, h * 4, 32);
            rank += (gh > gk) ? 1 : 0;
        }
        const bool sel = rank < 4;

        // --- sortable 32-bit keys: [f32 monotone, top 24b | (255 - idx)] ----
        // masked experts keep only the idx byte (< any valid key, never selected)
        const unsigned msel = sel ? 0xFFFFFF00u : 0u;
        unsigned k0, k1, k2, k3, k4, k5, k6, k7;
        #define MK(reg, j, v) { \
            unsigned _u = fsortable(v); \
            reg = (_u & msel) | ((laneoff + (unsigned)(j)) ^ 0xFFu); }
        MK(k0, 0, e0) MK(k1, 1, e1) MK(k2, 2, e2) MK(k3, 3, e3)
        MK(k4, 4, e4) MK(k5, 5, e5) MK(k6, 6, e6) MK(k7, 7, e7)
        #undef MK

        // --- Batcher odd-even sort of 8, descending (19 CE = min/max pairs) -
        CE(k0, k1) CE(k2, k3) CE(k4, k5) CE(k6, k7)
        CE(k0, k2) CE(k1, k3) CE(k4, k6) CE(k5, k7)
        CE(k1, k2) CE(k5, k6)
        CE(k0, k4) CE(k1, k5) CE(k2, k6) CE(k3, k7)
        CE(k2, k4) CE(k3, k5)
        CE(k1, k2) CE(k3, k4) CE(k5, k6)

        // --- extract global top-8: butterfly max of per-lane heads ----------
        int widx = 0;   // lane j (j<8) ends up holding the j-th expert index
        #pragma unroll
        for (int j = 0; j < 8; ++j) {
            unsigned m = k0;
            m = umax32(m, (unsigned)__shfl_xor((int)m, 16, 32));
            m = umax32(m, (unsigned)__shfl_xor((int)m,  8, 32));
            m = umax32(m, (unsigned)__shfl_xor((int)m,  4, 32));
            m = umax32(m, (unsigned)__shfl_xor((int)m,  2, 32));
            m = umax32(m, (unsigned)__shfl_xor((int)m,  1, 32));
            if (lane == j) widx = (int)((~m) & 0xFFu);
            const bool won = (k0 == m);     // keys unique: exactly one lane wins
            k0 = won ? k1 : k0;
            k1 = won ? k2 : k1;
            k2 = won ? k3 : k2;
            k3 = won ? k4 : k3;
            k4 = won ? k5 : k4;
            k5 = won ? k6 : k5;
            k6 = won ? k7 : k6;
            k7 = won ? 0u : k7;
        }

        // --- gather unbiased scores of winners, renormalize, scale ----------
        const int slot = lane & 7;
        const int idx8 = __shfl(widx, slot, 32);           // winner 'slot' for all lanes
        const float w = fast_sigmoid(x[((unsigned)t << 8) + (unsigned)idx8]); // cache hit
        float ssum = w;
        ssum += __shfl_xor(ssum, 1, 32);
        ssum += __shfl_xor(ssum, 2, 32);
        ssum += __shfl_xor(ssum, 4, 32);
        float dn = ssum + 1e-20f;
        float r = __builtin_amdgcn_rcpf(dn);
        r = __builtin_fmaf(__builtin_fmaf(-dn, r, 1.0f), r, r);
        const float yv = w * 2.5f * r;

        if (lane < 8) {
            const unsigned ooff = ((unsigned)t << 3) + (unsigned)lane;
            out_y[ooff]   = yv;
            out_idx[ooff] = widx;
        }
    }
}

extern "C" void kernel_launch(void* const* d_in, const int* in_sizes, int n_in,
                              void* d_out, int out_size, void* d_ws, size_t ws_size,
                              hipStream_t stream) {
    const float* x    = (const float*)d_in[0];
    const float* bias = (const float*)d_in[1];
    const int n = in_sizes[0] / 256;                 // tokens; 256 experts fixed

    float* out        = (float*)d_out;
    float* out_y      = out;                         // [n,8] f32
    int*   out_idx    = (int*)(out + (size_t)n * 8); // [n,8] i32 (bit-stored)
    float* out_scores = out + (size_t)n * 16;        // [n,256] f32

    const int waves  = (n + TPW - 1) / TPW;
    const int blocks = (waves + WAVES_PER_BLOCK - 1) / WAVES_PER_BLOCK;
    moe_gate_topk<<<blocks, 32 * WAVES_PER_BLOCK, 0, stream>>>(
        x, bias, out_y, out_idx, out_scores, n);
}